// AlignedBatchDynamicSoftLabelAssigner_86251533238822
// MI455X (gfx1250) — compile-verified
//
#include <hip/hip_runtime.h>

typedef __attribute__((ext_vector_type(2))) float v2f;
typedef __attribute__((ext_vector_type(8))) float v8f;

#define TOPK_K 13
#define BIG_INF 100000000.0f

__device__ __forceinline__ float sigf(float x) { return 1.0f / (1.0f + __expf(-x)); }

// ---------------------------------------------------------------------------
// K1: per (b,p): overlaps[b][g][p] for all g, valid_mask[b,p], zero atomics.
// ---------------------------------------------------------------------------
__global__ void k1_prep(const float* __restrict__ pred_bboxes,
                        const float* __restrict__ priors,
                        const float* __restrict__ gt_bboxes,
                        const float* __restrict__ pad_flag,
                        float* __restrict__ ov, float* __restrict__ valid,
                        unsigned* __restrict__ fg_pre, unsigned* __restrict__ mcnt,
                        unsigned* __restrict__ mgsum,
                        int P, int G) {
  int b = blockIdx.y;
  int p = blockIdx.x * blockDim.x + threadIdx.x;
  extern __shared__ float sh[];  // G*5 floats: box(4) + pad
  for (int i = threadIdx.x; i < G * 5; i += blockDim.x) {
    int g = i / 5, j = i % 5;
    sh[i] = (j < 4) ? gt_bboxes[(b * G + g) * 4 + j] : pad_flag[b * G + g];
  }
  __syncthreads();
  if (p >= P) return;
  float px = priors[p * 4 + 0], py = priors[p * 4 + 1];
  const float* pb = &pred_bboxes[((long)b * P + p) * 4];
  float q0 = pb[0], q1 = pb[1], q2 = pb[2], q3 = pb[3];
  float areaP = (q2 - q0) * (q3 - q1);
  int anyv = 0;
  for (int g = 0; g < G; ++g) {
    float gx0 = sh[g * 5 + 0], gy0 = sh[g * 5 + 1];
    float gx1 = sh[g * 5 + 2], gy1 = sh[g * 5 + 3], pad = sh[g * 5 + 4];
    float ming = fminf(fminf(px - gx0, py - gy0), fminf(gx1 - px, gy1 - py));
    anyv |= ((ming > 0.0f) && (pad > 0.0f));
    float areaG = (gx1 - gx0) * (gy1 - gy0);
    float w = fmaxf(fminf(gx1, q2) - fmaxf(gx0, q0), 0.0f);
    float h = fmaxf(fminf(gy1, q3) - fmaxf(gy0, q1), 0.0f);
    float inter = w * h;
    float uni = fmaxf(areaG + areaP - inter, 1e-6f);
    ov[((long)(b * G + g)) * P + p] = inter / uni;
  }
  long idx = (long)b * P + p;
  valid[idx] = anyv ? 1.0f : 0.0f;
  fg_pre[idx] = 0u; mcnt[idx] = 0u; mgsum[idx] = 0u;
}

// ---------------------------------------------------------------------------
// K2: one wave per 16x16 (p x g) tile. Logits via one-hot WMMA f32 16x16x4,
//     fused cost epilogue -> cost[b][p][g].
// ---------------------------------------------------------------------------
__global__ void k2_cost(const float* __restrict__ pred_scores,
                        const int* __restrict__ gt_labels,
                        const float* __restrict__ gt_bboxes,
                        const float* __restrict__ priors,
                        const float* __restrict__ ov,
                        const float* __restrict__ valid,
                        float* __restrict__ cost,
                        int P, int G, int C, int GT, int tilesPerB) {
  int lane = threadIdx.x & 31;
  int wave = threadIdx.x >> 5;
  int b = blockIdx.y;
  int tile = blockIdx.x * (blockDim.x >> 5) + wave;
  if (tile >= tilesPerB) return;
  int pt = tile / GT, gtile = tile % GT;
  int p0 = pt * 16, g0 = gtile * 16;
  int n = lane & 15, half = lane >> 4;
  int g = g0 + n;
  int lab = (g < G) ? gt_labels[b * G + g] : -1;
  int prow = p0 + n; if (prow > P - 1) prow = P - 1;
  const float* srow = pred_scores + ((long)b * P + prow) * C;
  v8f acc = {0.f, 0.f, 0.f, 0.f, 0.f, 0.f, 0.f, 0.f};
  for (int k0 = 0; k0 < C; k0 += 4) {
    int kk = k0 + 2 * half;
    v2f a; a.x = srow[kk]; a.y = srow[kk + 1];
    v2f bm; bm.x = (lab == kk) ? 1.0f : 0.0f; bm.y = (lab == kk + 1) ? 1.0f : 0.0f;
    acc = __builtin_amdgcn_wmma_f32_16x16x4_f32(false, a, false, bm,
                                                (short)0, acc, false, false);
  }
  if (g >= G) return;
  float gx0 = gt_bboxes[(b * G + g) * 4 + 0], gy0 = gt_bboxes[(b * G + g) * 4 + 1];
  float gx1 = gt_bboxes[(b * G + g) * 4 + 2], gy1 = gt_bboxes[(b * G + g) * 4 + 3];
  float cx = 0.5f * (gx0 + gx1), cy = 0.5f * (gy0 + gy1);
#pragma unroll
  for (int r = 0; r < 8; ++r) {
    int p = p0 + r + 8 * half;
    if (p >= P) continue;
    float logit = acc[r];
    float iou = ov[((long)(b * G + g)) * P + p];
    float px = priors[p * 4 + 0], py = priors[p * 4 + 1], st = priors[p * 4 + 2];
    float vm = valid[(long)b * P + p];
    float dx = px - cx, dy = py - cy;
    float dist = sqrtf(dx * dx + dy * dy) / st;
    dist *= vm;
    float soft = exp2f((dist - 3.0f) * 3.3219280948873623f);  // 10^(dist-3)
    float iou_cost = -__logf(iou + 1e-7f) * 3.0f;
    float sg = sigf(logit);
    float scale = iou - sg;
    float bce = fmaxf(logit, 0.0f) + log1pf(__expf(-fabsf(logit))) - logit * iou;
    float ctot = bce * scale * scale + iou_cost + soft;
    cost[((long)b * P + p) * G + g] = (vm > 0.0f) ? ctot : BIG_INF;
  }
}

// ---------------------------------------------------------------------------
// K3: per (b,g): top-13 of metrics over p; fg_pre += in_gts winners.
// ---------------------------------------------------------------------------
__global__ void k3_topk_metrics(const float* __restrict__ pred_scores,
                                const int* __restrict__ gt_labels,
                                const float* __restrict__ gt_bboxes,
                                const float* __restrict__ priors,
                                const float* __restrict__ pad_flag,
                                const float* __restrict__ ov,
                                unsigned* __restrict__ fg_pre,
                                int P, int G, int C) {
  int g = blockIdx.x, b = blockIdx.y;
  if (pad_flag[b * G + g] == 0.0f) return;
  extern __shared__ float vals[];  // P floats
  __shared__ float rv[256];
  __shared__ int ri[256];
  int lab = gt_labels[b * G + g];
  float gx0 = gt_bboxes[(b * G + g) * 4 + 0], gy0 = gt_bboxes[(b * G + g) * 4 + 1];
  float gx1 = gt_bboxes[(b * G + g) * 4 + 2], gy1 = gt_bboxes[(b * G + g) * 4 + 3];
  for (int p = threadIdx.x; p < P; p += blockDim.x) {
    float iou = ov[((long)(b * G + g)) * P + p];
    float s = pred_scores[((long)b * P + p) * C + lab];
    float px = priors[p * 4 + 0], py = priors[p * 4 + 1];
    float ming = fminf(fminf(px - gx0, py - gy0), fminf(gx1 - px, gy1 - py));
    float ing = (ming > 0.0f) ? 1.0f : 0.0f;
    float t = iou * iou * iou;
    vals[p] = sigf(s) * (t * t) * ing;
  }
  __syncthreads();
  for (int it = 0; it < TOPK_K; ++it) {
    float bv = -1.0f; int bi = P;
    for (int p = threadIdx.x; p < P; p += blockDim.x) {
      float v = vals[p];
      if (v > bv) { bv = v; bi = p; }
    }
    rv[threadIdx.x] = bv; ri[threadIdx.x] = bi;
    __syncthreads();
    for (int s = 128; s > 0; s >>= 1) {
      if (threadIdx.x < s) {
        float v2 = rv[threadIdx.x + s]; int i2 = ri[threadIdx.x + s];
        if (v2 > rv[threadIdx.x] || (v2 == rv[threadIdx.x] && i2 < ri[threadIdx.x])) {
          rv[threadIdx.x] = v2; ri[threadIdx.x] = i2;
        }
      }
      __syncthreads();
    }
    if (threadIdx.x == 0) {
      int sel = ri[0];
      float px = priors[sel * 4 + 0], py = priors[sel * 4 + 1];
      float ming = fminf(fminf(px - gx0, py - gy0), fminf(gx1 - px, gy1 - py));
      if (ming > 0.0f) atomicAdd(&fg_pre[(long)b * P + sel], 1u);
      vals[sel] = -2.0f;
    }
    __syncthreads();
  }
}

// ---------------------------------------------------------------------------
// K4: per (b,g): dynamic_k = max(trunc(sum(top13 iou)),1); select dynamic_k
//     smallest-cost priors; scatter via atomic count + g-sum.
// ---------------------------------------------------------------------------
__global__ void k4_select(const float* __restrict__ pad_flag,
                          const float* __restrict__ ov,
                          const float* __restrict__ cost,
                          unsigned* __restrict__ mcnt, unsigned* __restrict__ mgsum,
                          int P, int G) {
  int g = blockIdx.x, b = blockIdx.y;
  if (pad_flag[b * G + g] == 0.0f) return;
  extern __shared__ float vals[];
  __shared__ float rv[256];
  __shared__ int ri[256];
  // phase 1: sum of top-13 ious
  for (int p = threadIdx.x; p < P; p += blockDim.x)
    vals[p] = ov[((long)(b * G + g)) * P + p];
  __syncthreads();
  float ksum = 0.0f;
  for (int it = 0; it < TOPK_K; ++it) {
    float bv = -1.0f; int bi = P;
    for (int p = threadIdx.x; p < P; p += blockDim.x) {
      float v = vals[p];
      if (v > bv) { bv = v; bi = p; }
    }
    rv[threadIdx.x] = bv; ri[threadIdx.x] = bi;
    __syncthreads();
    for (int s = 128; s > 0; s >>= 1) {
      if (threadIdx.x < s) {
        float v2 = rv[threadIdx.x + s]; int i2 = ri[threadIdx.x + s];
        if (v2 > rv[threadIdx.x] || (v2 == rv[threadIdx.x] && i2 < ri[threadIdx.x])) {
          rv[threadIdx.x] = v2; ri[threadIdx.x] = i2;
        }
      }
      __syncthreads();
    }
    ksum += rv[0];
    if (threadIdx.x == 0) vals[ri[0]] = -2.0f;
    __syncthreads();
  }
  int dynk = (int)ksum; if (dynk < 1) dynk = 1;
  // phase 2: dynk smallest-cost priors for this gt
  for (int p = threadIdx.x; p < P; p += blockDim.x)
    vals[p] = cost[((long)b * P + p) * G + g];
  __syncthreads();
  for (int it = 0; it < dynk; ++it) {
    float bv = 3.0e38f; int bi = P;
    for (int p = threadIdx.x; p < P; p += blockDim.x) {
      float v = vals[p];
      if (v < bv) { bv = v; bi = p; }
    }
    rv[threadIdx.x] = bv; ri[threadIdx.x] = bi;
    __syncthreads();
    for (int s = 128; s > 0; s >>= 1) {
      if (threadIdx.x < s) {
        float v2 = rv[threadIdx.x + s]; int i2 = ri[threadIdx.x + s];
        if (v2 < rv[threadIdx.x] || (v2 == rv[threadIdx.x] && i2 < ri[threadIdx.x])) {
          rv[threadIdx.x] = v2; ri[threadIdx.x] = i2;
        }
      }
      __syncthreads();
    }
    if (threadIdx.x == 0) {
      int sel = ri[0];
      atomicAdd(&mcnt[(long)b * P + sel], 1u);
      atomicAdd(&mgsum[(long)b * P + sel], (unsigned)g);
      vals[sel] = 3.2e38f;
    }
    __syncthreads();
  }
}

// ---------------------------------------------------------------------------
// K5: finalize all 5 outputs (concatenated flat as float).
// ---------------------------------------------------------------------------
__global__ void k5_final(const float* __restrict__ cost, const float* __restrict__ ov,
                         const int* __restrict__ gt_labels,
                         const float* __restrict__ gt_bboxes,
                         const unsigned* __restrict__ fg_pre,
                         const unsigned* __restrict__ mcnt,
                         const unsigned* __restrict__ mgsum,
                         float* __restrict__ out,
                         int B, int P, int G, int NCLS) {
  int b = blockIdx.y;
  int p = blockIdx.x * blockDim.x + threadIdx.x;
  if (p >= P) return;
  long idx = (long)b * P + p;
  long NP = (long)B * P;
  unsigned c = mcnt[idx];
  int fg = (c > 0u);
  int gstar = 0;
  if (c == 1u) {
    gstar = (int)mgsum[idx];
  } else if (c > 1u) {
    const float* crow = &cost[idx * G];
    float bv = crow[0];
    for (int g = 1; g < G; ++g) {
      float v = crow[g];
      if (v < bv) { bv = v; gstar = g; }
    }
  }
  out[idx] = fg ? (float)gt_labels[b * G + gstar] : (float)NCLS;
  out[NP + idx] = 1.0f;
  float b0 = 0.f, b1 = 0.f, b2 = 0.f, b3 = 0.f, mi = 0.f;
  if (fg) {
    const float* gb = &gt_bboxes[(b * G + gstar) * 4];
    b0 = gb[0]; b1 = gb[1]; b2 = gb[2]; b3 = gb[3];
    mi = ov[((long)(b * G + gstar)) * P + p];
  }
  out[2 * NP + idx * 4 + 0] = b0;
  out[2 * NP + idx * 4 + 1] = b1;
  out[2 * NP + idx * 4 + 2] = b2;
  out[2 * NP + idx * 4 + 3] = b3;
  out[6 * NP + idx] = mi;
  out[7 * NP + idx] = (fg_pre[idx] > 0u) ? 1.0f : 0.0f;
}

static inline size_t align256(size_t x) { return (x + 255) & ~(size_t)255; }

extern "C" void kernel_launch(void* const* d_in, const int* in_sizes, int n_in,
                              void* d_out, int out_size, void* d_ws, size_t ws_size,
                              hipStream_t stream) {
  const float* pred_bboxes = (const float*)d_in[0];
  const float* pred_scores = (const float*)d_in[1];
  const float* priors      = (const float*)d_in[2];
  const int*   gt_labels   = (const int*)d_in[3];
  const float* gt_bboxes   = (const float*)d_in[4];
  const float* pad_flag    = (const float*)d_in[5];

  int P = in_sizes[2] / 4;
  int B = in_sizes[0] / (P * 4);
  int G = in_sizes[4] / (B * 4);
  int C = in_sizes[1] / (B * P);

  char* ws = (char*)d_ws;
  size_t szOv   = (size_t)B * G * P * sizeof(float);
  size_t szCost = (size_t)B * P * G * sizeof(float);
  size_t szBP   = (size_t)B * P * sizeof(unsigned);
  size_t o = 0;
  float*    ov     = (float*)(ws + o);    o = align256(o + szOv);
  float*    cost   = (float*)(ws + o);    o = align256(o + szCost);
  float*    valid  = (float*)(ws + o);    o = align256(o + szBP);
  unsigned* fg_pre = (unsigned*)(ws + o); o = align256(o + szBP);
  unsigned* mcnt   = (unsigned*)(ws + o); o = align256(o + szBP);
  unsigned* mgsum  = (unsigned*)(ws + o); o = align256(o + szBP);
  (void)ws_size;

  float* out = (float*)d_out;

  dim3 blk(256);
  dim3 grdP((P + 255) / 256, B);
  size_t shK1 = (size_t)G * 5 * sizeof(float);
  k1_prep<<<grdP, blk, shK1, stream>>>(pred_bboxes, priors, gt_bboxes, pad_flag,
                                       ov, valid, fg_pre, mcnt, mgsum, P, G);

  int PT = (P + 15) / 16, GT = (G + 15) / 16;
  int tilesPerB = PT * GT;
  dim3 grdT((tilesPerB + 7) / 8, B);  // 8 waves (256 threads) per block
  k2_cost<<<grdT, blk, 0, stream>>>(pred_scores, gt_labels, gt_bboxes, priors,
                                    ov, valid, cost, P, G, C, GT, tilesPerB);

  size_t shRow = (size_t)P * sizeof(float);
  dim3 grdG(G, B);
  k3_topk_metrics<<<grdG, blk, shRow, stream>>>(pred_scores, gt_labels, gt_bboxes,
                                                priors, pad_flag, ov, fg_pre, P, G, C);

  k4_select<<<grdG, blk, shRow, stream>>>(pad_flag, ov, cost, mcnt, mgsum, P, G);

  k5_final<<<grdP, blk, 0, stream>>>(cost, ov, gt_labels, gt_bboxes, fg_pre,
                                     mcnt, mgsum, out, B, P, G, 80);
}